// WeightedWassersteinLoss_4088808866113
// MI455X (gfx1250) — compile-verified
//
#include <hip/hip_runtime.h>
#include <math.h>

// -----------------------------------------------------------------------------
// WeightedWassersteinLoss (debiased Sinkhorn divergence, 1-D clouds per dim)
// MI455X / gfx1250 design:
//   * cost tile s_ij = A_j + c_i*y_j + d_i as a K=4 f32 WMMA:
//       A row i = [c_i, d_i, 1, 0],  B col j = [y_j, 1, A_j, 0]^T
//     -> 1024 FMAs per v_wmma_f32_16x16x4_f32 on the matrix pipe, leaving
//        VALU/TRANS free for max / v_exp_f32 (co-execution)
//   * pass 2 feeds C = -rowmax into the WMMA accumulator: the numerically
//     stabilizing subtraction rides the matrix pipe for free
//   * row reductions via single-instruction ds_swizzle_b32 XOR butterflies
//     (SWAPX1/2/4/8) confined to 16-lane halves (wave32 D-tile layout)
//   * column cloud + log-weights staged in LDS (16 KB), broadcast reads
//   * epsilon annealing = dependent sequence -> ~55 stream launches
// -----------------------------------------------------------------------------

#define B_N 2048
#define D_N 8
#define NEG_LW -1e9f

typedef __attribute__((ext_vector_type(2))) float v2f;
typedef __attribute__((ext_vector_type(8))) float v8f;

// single-instruction XOR butterfly within 16-lane halves (group-of-32 swizzle,
// offset = {1'b0, and=0x1F, or=0, xor=mask} -> src_lane = lane ^ mask)
#define SWZ_XOR(v, imm) __int_as_float(__builtin_amdgcn_ds_swizzle(__float_as_int(v), (imm)))

// ---------------------------------------------------------------- prep -------
// one block per dim: transpose preds/targets to [D][B], build mask, counts,
// masked log-weights  logw = -log(n_valid) (valid) / -1e9 (NaN target)
__global__ void ww_prep_kernel(const float* __restrict__ preds,
                               const float* __restrict__ targ,
                               float* __restrict__ Xt, float* __restrict__ Yt,
                               float* __restrict__ LW, float* __restrict__ counts)
{
    const int d = blockIdx.x;
    const int tid = threadIdx.x;
    __shared__ int red[256];
    __shared__ float s_logn;

    int cnt = 0;
    for (int i = tid; i < B_N; i += 256) {
        float t = targ[i * D_N + d];
        bool valid = !(t != t);
        cnt += valid ? 1 : 0;
        Yt[d * B_N + i] = valid ? t : 0.0f;
        Xt[d * B_N + i] = preds[i * D_N + d];
    }
    red[tid] = cnt;
    __syncthreads();
    for (int s = 128; s > 0; s >>= 1) {
        if (tid < s) red[tid] += red[tid + s];
        __syncthreads();
    }
    if (tid == 0) {
        counts[d] = (float)red[0];
        int c = red[0] > 1 ? red[0] : 1;
        s_logn = __logf((float)c);
    }
    __syncthreads();
    const float logn = s_logn;
    for (int i = tid; i < B_N; i += 256) {
        float t = targ[i * D_N + d];
        LW[d * B_N + i] = (t != t) ? NEG_LW : -logn;
    }
}

// ------------------------------------------------------------- softmin -------
// grid = (B/64 row-tiles, 4 jobs, D dims), block = 128 (4 waves, 16 rows each)
// job 0: f  <- softmin(Cxy , h=lw[+g/eps]) ; job 1: g <- softmin(Cyx, h=lw[+f/eps])
// job 2: a  <- softmin(Cxx , h=lw[+a/eps]) ; job 3: b <- softmin(Cyy, h=lw[+b/eps])
// damp_mask bit per job: out = 0.5*(old + val), else out = val.
__global__ __launch_bounds__(128)
void ww_softmin_kernel(const float* __restrict__ X, const float* __restrict__ Y,
                       const float* __restrict__ LW,
                       const float* __restrict__ Fin, const float* __restrict__ Gin,
                       const float* __restrict__ Ain, const float* __restrict__ Bin,
                       float* __restrict__ Fout, float* __restrict__ Gout,
                       float* __restrict__ Aout, float* __restrict__ Bout,
                       float eps, int use_dual, int damp_mask)
{
    __shared__ float sh_col[B_N];   // column cloud points
    __shared__ float sh_h[B_N];     // A_j = lw_j + dual_j/eps - col_j^2/(2 eps)

    const int d = blockIdx.z;
    const int job = (int)blockIdx.y;
    const size_t o = (size_t)d * B_N;
    const float inv_eps = 1.0f / eps;

    const float *rows, *cols, *dual, *oldv;
    float* outv;
    switch (job) {
      case 0:  rows = X + o; cols = Y + o; dual = Gin + o; oldv = Fin + o; outv = Fout + o; break;
      case 1:  rows = Y + o; cols = X + o; dual = Fin + o; oldv = Gin + o; outv = Gout + o; break;
      case 2:  rows = X + o; cols = X + o; dual = Ain + o; oldv = Ain + o; outv = Aout + o; break;
      default: rows = Y + o; cols = Y + o; dual = Bin + o; oldv = Bin + o; outv = Bout + o; break;
    }
    const bool damp = (damp_mask >> job) & 1;
    const float* lw = LW + o;

    for (int j = threadIdx.x; j < B_N; j += 128) {
        float cv = cols[j];
        float h = lw[j];
        if (use_dual) h += dual[j] * inv_eps;
        sh_col[j] = cv;
        sh_h[j]   = h - 0.5f * inv_eps * cv * cv;
    }
    __syncthreads();

    const int lane = threadIdx.x & 31;
    const int wave = threadIdx.x >> 5;
    const int half = lane >> 4;        // 0: lanes 0-15, 1: lanes 16-31
    const int n16  = lane & 15;
    const int r0 = (blockIdx.x * 4 + wave) * 16;

    // A matrix 16x4: row i = [c_i, d_i, 1, 0]
    // layout: lanes 0-15 hold (K0,K1)=(c,d) of row n16; lanes 16-31 hold (K2,K3)=(1,0)
    const float rp = rows[r0 + n16];
    const float ci = rp * inv_eps;
    const float di = -0.5f * inv_eps * rp * rp;
    v2f Amat;
    Amat.x = half ? 1.0f : ci;
    Amat.y = half ? 0.0f : di;

    // B matrix 4x16: col j = [y_j, 1, A_j, 0]^T
    // layout: VGPR0 lanes0-15 = K0 row (y_j), lanes16-31 = K2 row (A_j)
    //         VGPR1 lanes0-15 = K1 row (1),   lanes16-31 = K3 row (0)
    const float* bsrc = half ? sh_h : sh_col;
    const float  bcon = half ? 0.0f : 1.0f;

    v8f zero = {0.f, 0.f, 0.f, 0.f, 0.f, 0.f, 0.f, 0.f};

    // ---- pass 1: row maxima (matrix pipe builds tiles, VALU takes max) ----
    float m[8];
    #pragma unroll
    for (int k = 0; k < 8; ++k) m[k] = -3.0e38f;

    #pragma unroll 4
    for (int jt = 0; jt < B_N; jt += 16) {
        v2f Bmat;
        Bmat.x = bsrc[jt + n16];
        Bmat.y = bcon;
        v8f t = __builtin_amdgcn_wmma_f32_16x16x4_f32(false, Amat, false, Bmat,
                                                      (short)0, zero, false, false);
        #pragma unroll
        for (int k = 0; k < 8; ++k) m[k] = fmaxf(m[k], t[k]);
    }
    // single-instruction butterfly max within each 16-lane half (rows k / k+8)
    #pragma unroll
    for (int k = 0; k < 8; ++k) {
        m[k] = fmaxf(m[k], SWZ_XOR(m[k], 0x7C01));
        m[k] = fmaxf(m[k], SWZ_XOR(m[k], 0x7C02));
        m[k] = fmaxf(m[k], SWZ_XOR(m[k], 0x7C04));
        m[k] = fmaxf(m[k], SWZ_XOR(m[k], 0x7C08));
    }

    // ---- pass 2: sum exp(t - m); the -m subtraction is folded into the WMMA
    //      accumulator (C layout == D layout: VGPR k <-> rows k / k+8) ----
    v8f negm;
    #pragma unroll
    for (int k = 0; k < 8; ++k) negm[k] = -m[k];

    float s[8];
    #pragma unroll
    for (int k = 0; k < 8; ++k) s[k] = 0.0f;

    #pragma unroll 4
    for (int jt = 0; jt < B_N; jt += 16) {
        v2f Bmat;
        Bmat.x = bsrc[jt + n16];
        Bmat.y = bcon;
        v8f t = __builtin_amdgcn_wmma_f32_16x16x4_f32(false, Amat, false, Bmat,
                                                      (short)0, negm, false, false);
        #pragma unroll
        for (int k = 0; k < 8; ++k) s[k] += __expf(t[k]);
    }
    #pragma unroll
    for (int k = 0; k < 8; ++k) {
        s[k] += SWZ_XOR(s[k], 0x7C01);
        s[k] += SWZ_XOR(s[k], 0x7C02);
        s[k] += SWZ_XOR(s[k], 0x7C04);
        s[k] += SWZ_XOR(s[k], 0x7C08);
    }

    // softmin value for row (r0 + half*8 + k): -eps*(m + log s)
    float val = 0.0f;
    #pragma unroll
    for (int k = 0; k < 8; ++k)
        if (n16 == k) val = -eps * (m[k] + __logf(s[k]));

    if (n16 < 8) {
        const int row = r0 + (half << 3) + n16;
        outv[row] = damp ? 0.5f * (oldv[row] + val) : val;
    }
    (void)zero;
}

// -------------------------------------------------------------- reduce ------
// per_dim[d] = sum_i w_i (f-a+g-b), zeroed when count<=1; out[0]=mean, out[1..8]=per_dim
__global__ void ww_reduce_kernel(const float* __restrict__ LW,
                                 const float* __restrict__ counts,
                                 const float* __restrict__ F, const float* __restrict__ G,
                                 const float* __restrict__ Aa, const float* __restrict__ Bb,
                                 float* __restrict__ out)
{
    __shared__ float red[256];
    const int tid = threadIdx.x;
    float total = 0.0f;
    for (int d = 0; d < D_N; ++d) {
        const size_t o = (size_t)d * B_N;
        float local = 0.0f;
        for (int i = tid; i < B_N; i += 256) {
            float w = __expf(LW[o + i]);          // 1/n valid, 0 masked
            local += w * ((F[o + i] - Aa[o + i]) + (G[o + i] - Bb[o + i]));
        }
        red[tid] = local;
        __syncthreads();
        for (int s = 128; s > 0; s >>= 1) {
            if (tid < s) red[tid] += red[tid + s];
            __syncthreads();
        }
        float pd = (counts[d] > 1.0f) ? red[0] : 0.0f;
        if (tid == 0) out[1 + d] = pd;
        total += pd;
        __syncthreads();
    }
    if (tid == 0) out[0] = total / (float)D_N;
}

// ---------------------------------------------------------------- host ------
extern "C" void kernel_launch(void* const* d_in, const int* in_sizes, int n_in,
                              void* d_out, int out_size, void* d_ws, size_t ws_size,
                              hipStream_t stream)
{
    (void)in_sizes; (void)n_in; (void)out_size; (void)ws_size;
    const float* preds = (const float*)d_in[0];
    const float* targ  = (const float*)d_in[1];
    float* out = (float*)d_out;

    // workspace layout (floats): Xt[N] Yt[N] LW[N] counts[16] duals0[4N] duals1[4N]
    const size_t N = (size_t)B_N * D_N;
    float* ws = (float*)d_ws;
    float* Xt = ws;
    float* Yt = Xt + N;
    float* LW = Yt + N;
    float* counts = LW + N;
    float* duals0 = counts + 16;
    float* duals1 = duals0 + 4 * N;

    // geomloss epsilon schedule: [diam^p] ++ exp(arange(p*log diam, p*log blur, p*log scal)) ++ [blur^p]
    float eps[64]; int ne = 0;
    {
        const double P = 2.0, BLUR = 0.05, SCALING = 0.9, DIA = 8.0;
        eps[ne++] = (float)pow(DIA, P);
        const double start = P * log(DIA), stop = P * log(BLUR), step = P * log(SCALING);
        const int cnt = (int)ceil((stop - start) / step);
        for (int i = 0; i < cnt; ++i) eps[ne++] = (float)exp(start + (double)i * step);
        eps[ne++] = (float)pow(BLUR, P);
    }

    ww_prep_kernel<<<D_N, 256, 0, stream>>>(preds, targ, Xt, Yt, LW, counts);

    const dim3 grid(B_N / 64, 4, D_N);
    const dim3 blk(128);

    float* cur = duals0;
    float* nxt = duals1;

    // init at eps[0] with h = logw (no duals, no damping)
    ww_softmin_kernel<<<grid, blk, 0, stream>>>(Xt, Yt, LW,
        cur, cur + N, cur + 2 * N, cur + 3 * N,
        cur, cur + N, cur + 2 * N, cur + 3 * N,
        eps[0], 0, 0);

    // damped Sinkhorn scan over the full schedule
    for (int t = 0; t < ne; ++t) {
        ww_softmin_kernel<<<grid, blk, 0, stream>>>(Xt, Yt, LW,
            cur, cur + N, cur + 2 * N, cur + 3 * N,
            nxt, nxt + N, nxt + 2 * N, nxt + 3 * N,
            eps[t], 1, 0xF);
        float* tmp = cur; cur = nxt; nxt = tmp;
    }

    // final extrapolation at last eps: f,g full updates; a,b damped
    ww_softmin_kernel<<<grid, blk, 0, stream>>>(Xt, Yt, LW,
        cur, cur + N, cur + 2 * N, cur + 3 * N,
        nxt, nxt + N, nxt + 2 * N, nxt + 3 * N,
        eps[ne - 1], 1, 0xC);
    cur = nxt;

    ww_reduce_kernel<<<1, 256, 0, stream>>>(LW, counts,
        cur, cur + N, cur + 2 * N, cur + 3 * N, out);
}